// Model6_12171937317370
// MI455X (gfx1250) — compile-verified
//
#include <hip/hip_runtime.h>
#include <stdint.h>

// ---------------------------------------------------------------------------
// MI455X / gfx1250. fp32 end-to-end (matches reference numerics) on the WMMA
// pipe via V_WMMA_F32_16X16X4_F32. Workload ~371 GFLOP fp32, HBM-bound at
// 23.3 TB/s, so f32 WMMA costs nothing vs bf16 and preserves precision.
// Round-3 changes (FC GEMM only):
//  * BUGFIX: k-chunk 48 (divides 3600 exactly; the old 32 over-ran each row
//    by 16 columns on the last chunk and accumulated garbage).
//  * Double-buffered LDS ping-pong: async copies for chunk t+1 issue before
//    computing chunk t, completed with s_wait_asynccnt 12 (ASYNCcnt is
//    in-order), so HBM fetch overlaps the 48 WMMAs per chunk.
// ---------------------------------------------------------------------------

typedef __attribute__((ext_vector_type(2))) float v2f;
typedef __attribute__((ext_vector_type(8))) float v8f;

__device__ __forceinline__ v8f wmma_f32_4(v2f a, v2f b, v8f c) {
    // (neg_a, A, neg_b, B, c_mod, C, reuse_a, reuse_b)
    return __builtin_amdgcn_wmma_f32_16x16x4_f32(false, a, false, b,
                                                 (short)0, c, false, false);
}

// gfx1250 async global->LDS copy (16B per lane), tracked by ASYNCcnt.
__device__ __forceinline__ void async_copy_f4(const float* gaddr, uint32_t lds_byte_addr) {
    asm volatile("global_load_async_to_lds_b128 %0, %1, off"
                 :: "v"(lds_byte_addr), "v"(gaddr) : "memory");
}
template <int N>
__device__ __forceinline__ void wait_async_le() {
    asm volatile("s_wait_asynccnt %0" :: "i"(N) : "memory");
}

#define BATCH 512
#define FCN   3600
#define MIRROR_C1_OFFSET (-1.2432432432432432f)

#define FC_KC  48           // k-chunk: 3600 = 75 * 48 (exact)
#define FC_LDW 52           // padded LDS row stride (floats): 208B, 16B-aligned,
                            // gcd(52,64)=4 -> spread over 16 bank groups
#define FC_NCH (FCN / FC_KC)        // 75 chunks
#define FC_BUFB (64 * FC_LDW * 4)   // bytes per LDS buffer (one matrix)

// ---------------------------------------------------------------------------
// FC layer: H = relu(X @ W^T + bias).  X:(512,3600) W:(3600,3600) H:(512,3600)
// Block = 128 threads = 4 waves. Block tile 64(M) x 64(N); wave w owns rows
// [w*16, w*16+16) x all 64 N (4 v8f accumulators). K chunked by 48 through
// double-buffered LDS filled by async global->LDS b128 copies.
// ---------------------------------------------------------------------------
__global__ __launch_bounds__(128)
void fc_gemm_relu(const float* __restrict__ X, const float* __restrict__ W,
                  const float* __restrict__ bias, float* __restrict__ H)
{
    __shared__ float As[2][64 * FC_LDW];
    __shared__ float Bs[2][64 * FC_LDW];

    const int tid  = threadIdx.x;
    const int lane = tid & 31;
    const int wave = tid >> 5;
    const int mblk = blockIdx.x * 64;   // 512/64 = 8 blocks in M
    const int nblk = blockIdx.y * 64;   // ceil(3600/64) = 57 blocks in N

    const int mrow = lane & 15;         // A: M index of this lane
    const int koff = (lane >> 4) * 2;   // A/B: K offset of this lane half
    const int ncol = lane & 15;         // B/C/D: N index of this lane

    // Cooperative staging: 64 rows x 12 float4-chunks = 768 units,
    // 128 threads -> 6 units per thread per matrix.
    const float* xptr[6];
    const float* wptr[6];
    uint32_t     aoff[6], boff[6];
#pragma unroll
    for (int u = 0; u < 6; ++u) {
        const int unit = tid + u * 128;
        const int r = unit / 12;            // 0..63
        const int c = (unit % 12) * 4;      // 0..44
        xptr[u] = X + (size_t)(mblk + r) * FCN + c;          // mblk+r < 512 always
        const int n = nblk + r;                              // clamp OOB N rows:
        wptr[u] = W + (size_t)(n < FCN ? n : FCN - 1) * FCN + c; // garbage stays in
        aoff[u] = (uint32_t)(uintptr_t)&As[0][r * FC_LDW + c];   // unstored columns
        boff[u] = (uint32_t)(uintptr_t)&Bs[0][r * FC_LDW + c];
    }

    auto issue_chunk = [&](int k0, int buf) {
        const uint32_t bo = (uint32_t)buf * FC_BUFB;
#pragma unroll
        for (int u = 0; u < 6; ++u) {
            async_copy_f4(xptr[u] + k0, aoff[u] + bo);
            async_copy_f4(wptr[u] + k0, boff[u] + bo);
        }
    };

    v8f acc0 = {}, acc1 = {}, acc2 = {}, acc3 = {};

    issue_chunk(0, 0);
    for (int t = 0; t < FC_NCH; ++t) {
        const int cur = t & 1;
        if (t + 1 < FC_NCH) {
            // Buffer cur^1 is free: the trailing barrier of iteration t-1
            // guarantees every wave finished reading it.
            issue_chunk((t + 1) * FC_KC, cur ^ 1);
            wait_async_le<12>();    // chunk t's 12 copies done; t+1's in flight
        } else {
            wait_async_le<0>();
        }
        __syncthreads();

#pragma unroll
        for (int k4 = 0; k4 < FC_KC; k4 += 4) {
            const v2f a  = *(const v2f*)&As[cur][(wave * 16 + mrow) * FC_LDW + k4 + koff];
            const v2f b0 = *(const v2f*)&Bs[cur][( 0 + ncol) * FC_LDW + k4 + koff];
            const v2f b1 = *(const v2f*)&Bs[cur][(16 + ncol) * FC_LDW + k4 + koff];
            const v2f b2 = *(const v2f*)&Bs[cur][(32 + ncol) * FC_LDW + k4 + koff];
            const v2f b3 = *(const v2f*)&Bs[cur][(48 + ncol) * FC_LDW + k4 + koff];
            acc0 = wmma_f32_4(a, b0, acc0);
            acc1 = wmma_f32_4(a, b1, acc1);
            acc2 = wmma_f32_4(a, b2, acc2);
            acc3 = wmma_f32_4(a, b3, acc3);
        }
        __syncthreads();
    }

    // C/D layout: VGPR j -> M = j (lanes 0-15) or j+8 (lanes 16-31); N = lane&15
    const int mhi = (lane >> 4) << 3;
#pragma unroll
    for (int j = 0; j < 8; ++j) {
        const int m = mblk + wave * 16 + j + mhi;
#pragma unroll
        for (int f = 0; f < 4; ++f) {
            const int n = nblk + f * 16 + ncol;
            if (n < FCN) {
                float v = (f == 0 ? acc0[j] : f == 1 ? acc1[j]
                                   : f == 2 ? acc2[j] : acc3[j]);
                v += bias[n];
                H[(size_t)m * FCN + n] = fmaxf(v, 0.f);
            }
        }
    }
}

// ---------------------------------------------------------------------------
// build_padded: (512,3,92,92) canvas with custom mirror padding.
// Column map first (left mirror w/ channel-1 sign+offset, right mirror), then
// row map (top rows -> 0 pre-offset, bottom mirror) — matches the reference's
// update ordering exactly, including the OFFSET band at top-left of channel 1.
// ---------------------------------------------------------------------------
__global__ __launch_bounds__(256)
void build_canvas(const float* __restrict__ x, const float* __restrict__ h,
                  float* __restrict__ canvas)
{
    const int total = BATCH * 3 * 92 * 92;
    int idx = blockIdx.x * 256 + threadIdx.x;
    if (idx >= total) return;

    int j = idx % 92;
    int t = idx / 92;
    int i = t % 92; t /= 92;
    int c = t % 3;
    int b = t / 3;

    float sign = 1.f, add = 0.f;
    int jj = j;
    if (j < 16) {                       // left mirror across col 16
        jj = 31 - j;
        if (c == 1) { sign = -1.f; add = MIRROR_C1_OFFSET; }
    }
    if (jj >= 76) jj = 151 - jj;        // right mirror of cols 60:76

    float v = 0.f;
    if (i >= 16) {                      // rows < 16 are zero pre-left-mirror
        int ii = (i >= 76) ? 151 - i : i;   // bottom mirror of rows 60:76
        int r = ii - 16, cc = jj - 16;
        if (c == 0)      v = x[(size_t)b * 3600 + r * 60 + cc];
        else if (c == 1) v = h[(size_t)b * 3600 + r * 60 + cc];
        else             v = (cc >= 51) ? 1.f : (cc >= 41 ? 0.5f : 0.f);
    }
    canvas[idx] = add + sign * v;
}

// ---------------------------------------------------------------------------
// 9x9 VALID conv as implicit GEMM on the WMMA pipe.
// M = 16 flattened output pixels (tail masked), N = 16 (Cout zero-padded in
// LDS), K = CIN*81 padded to multiple of 4. One wave per (batch, pixel tile).
// Weights live in LDS; the im2col A-gather is pipelined one k-step ahead.
// ---------------------------------------------------------------------------
template <int CIN, int COUT, bool RELU>
__global__ __launch_bounds__(256)
void conv9_wmma(const float* __restrict__ in, const float* __restrict__ w,
                const float* __restrict__ bias, float* __restrict__ out,
                int IH, int IW, int total_waves)
{
    constexpr int K  = CIN * 81;
    constexpr int K4 = (K + 3) & ~3;
    __shared__ float Ws[16 * K4];       // [n][k], rows >= COUT and k >= K are 0

    const int tid = threadIdx.x;
    for (int idx = tid; idx < 16 * K4; idx += 256) {
        const int n = idx / K4, k = idx - n * K4;
        Ws[idx] = (n < COUT && k < K) ? w[n * K + k] : 0.f;
    }
    __syncthreads();    // all threads reach this before any wave may exit

    const int OH = IH - 8, OW = IW - 8;
    const int lane = tid & 31;
    const int wid = ((blockIdx.x * blockDim.x + tid) >> 5); // uniform per wave
    if (wid >= total_waves) return;     // wave-uniform: EXEC stays full for WMMA

    const int npix = OH * OW;
    const int tiles_per_img = (npix + 15) >> 4;
    const int b  = wid / tiles_per_img;
    const int p0 = (wid % tiles_per_img) * 16;

    const int mrow = lane & 15;
    const int koff = (lane >> 4) * 2;
    const int ncol = lane & 15;

    const int p = p0 + mrow;
    const bool pvalid = p < npix;
    const int oy = pvalid ? p / OW : 0;
    const int ox = pvalid ? p % OW : 0;
    const float* __restrict__ inb = in + (size_t)b * CIN * IH * IW;

    // im2col fragment gather for reduction indices [k0+koff, k0+koff+1]
    auto loadA = [&](int k0) -> v2f {
        v2f a;
        const int ka = k0 + koff;
        float v0 = 0.f, v1 = 0.f;
        if (pvalid && ka < K) {
            const int cc = ka / 81, rr = (ka % 81) / 9, ss = ka % 9; // const divides
            v0 = inb[(size_t)cc * IH * IW + (oy + rr) * IW + (ox + ss)];
        }
        if (pvalid && (ka + 1) < K) {
            const int k1 = ka + 1;
            const int cc = k1 / 81, rr = (k1 % 81) / 9, ss = k1 % 9;
            v1 = inb[(size_t)cc * IH * IW + (oy + rr) * IW + (ox + ss)];
        }
        a.x = v0; a.y = v1;
        return a;
    };

    const float* __restrict__ wr = &Ws[ncol * K4];

    v8f acc = {};
    v2f a_cur = loadA(0);
    for (int k0 = 0; k0 < K4; k0 += 4) {
        v2f a_nxt = {};
        if (k0 + 4 < K4) a_nxt = loadA(k0 + 4);         // prefetch next step
        const v2f bf = *(const v2f*)&wr[k0 + koff];      // LDS, zero-padded
        acc = wmma_f32_4(a_cur, bf, acc);
        a_cur = a_nxt;
    }

    const float bi = (ncol < COUT) ? bias[ncol] : 0.f;
    float* __restrict__ outb = out + (size_t)b * COUT * npix;
    const int mhi = (lane >> 4) << 3;
#pragma unroll
    for (int j = 0; j < 8; ++j) {
        const int pp = p0 + j + mhi;
        if (ncol < COUT && pp < npix) {
            float v = acc[j] + bi;
            if (RELU) v = fmaxf(v, 0.f);
            outb[(size_t)ncol * npix + pp] = v;
        }
    }
}

// ---------------------------------------------------------------------------
// Host: FC + 5 powerlayers on `stream`.
// Workspace layout (floats):
//   [canvas | c3out]  max(512*3*92*92, 512*8*68*68) = 18,939,904
//   c1out             512*8*84*84                   = 28,901,376
//   c2out             512*8*76*76                   = 23,658,496
//   hA, hB            512*3600 each                 =  1,843,200 x2
// total ~ 301 MB of d_ws.
// ---------------------------------------------------------------------------
extern "C" void kernel_launch(void* const* d_in, const int* in_sizes, int n_in,
                              void* d_out, int out_size, void* d_ws, size_t ws_size,
                              hipStream_t stream)
{
    const float* x    = (const float*)d_in[0];
    const float* fc_w = (const float*)d_in[1];
    const float* fc_b = (const float*)d_in[2];
    const float* w1   = (const float*)d_in[3];
    const float* bC1  = (const float*)d_in[4];
    const float* w2   = (const float*)d_in[5];
    const float* bC2  = (const float*)d_in[6];
    const float* w3   = (const float*)d_in[7];
    const float* bC3  = (const float*)d_in[8];
    const float* w4   = (const float*)d_in[9];
    const float* bC4  = (const float*)d_in[10];
    float* out = (float*)d_out;

    float* ws = (float*)d_ws;
    float* canvas = ws;                         // reused as c3out after conv1
    float* c3out  = canvas;
    float* c1out  = canvas + 18939904;
    float* c2out  = c1out + 28901376;
    float* hA     = c2out + 23658496;
    float* hB     = hA + BATCH * FCN;

    // FC: h = relu(X @ W^T + b)
    {
        dim3 grid(8, 57), block(128);
        fc_gemm_relu<<<grid, block, 0, stream>>>(x, fc_w, fc_b, hA);
    }

    const float* hin = hA;
    for (int it = 0; it < 5; ++it) {
        float* hout = (it == 4) ? out : ((it & 1) ? hA : hB);

        build_canvas<<<(BATCH * 3 * 92 * 92) / 256, 256, 0, stream>>>(x, hin, canvas);

        // conv1: (B,3,92,92) -> (B,8,84,84)
        { int waves = BATCH * ((84 * 84 + 15) / 16);
          conv9_wmma<3, 8, true><<<(waves + 7) / 8, 256, 0, stream>>>(
              canvas, w1, bC1, c1out, 92, 92, waves); }
        // conv2: (B,8,84,84) -> (B,8,76,76)
        { int waves = BATCH * ((76 * 76 + 15) / 16);
          conv9_wmma<8, 8, true><<<(waves + 7) / 8, 256, 0, stream>>>(
              c1out, w2, bC2, c2out, 84, 84, waves); }
        // conv3: (B,8,76,76) -> (B,8,68,68)   (overwrites dead canvas)
        { int waves = BATCH * ((68 * 68 + 15) / 16);
          conv9_wmma<8, 8, true><<<(waves + 7) / 8, 256, 0, stream>>>(
              c2out, w3, bC3, c3out, 76, 76, waves); }
        // conv4: (B,8,68,68) -> (B,1,60,60), no relu
        { int waves = BATCH * ((60 * 60 + 15) / 16);
          conv9_wmma<8, 1, false><<<(waves + 7) / 8, 256, 0, stream>>>(
              c3out, w4, bC4, hout, 68, 68, waves); }

        hin = hout;
    }
}